// MVR_MS_reg_32238024524416
// MI455X (gfx1250) — compile-verified
//
#include <hip/hip_runtime.h>
#include <math.h>

typedef __attribute__((ext_vector_type(2))) float v2f;
typedef __attribute__((ext_vector_type(8))) float v8f;

#define B_N 4096
#define D_N 128
#define NT  (B_N / 16)        // 256 column tiles
#define WAVES 8               // waves per block in tile kernels
#define TPW  (NT / WAVES)     // 32 column tiles per wave
#define BIGF 1.0e30f

#define SCALE_POS 2.0f
#define SCALE_NEG 50.0f
#define THRESH    0.7f
#define MARGIN    0.1f
#define GAMMA     0.3f
#define ONE_M_EPS 0.99999f    // 1 - 1e-5
#define COS_EPS   1e-8f

__device__ __forceinline__ v8f wmma_f32(v2f a, v2f b, v8f c) {
  return __builtin_amdgcn_wmma_f32_16x16x4_f32(
      /*neg_a=*/false, a, /*neg_b=*/false, b,
      /*c_mod=*/(short)0, c, /*reuse_a=*/false, /*reuse_b=*/false);
}

// B-fragment group loaders / consumers. Distinct buffer names keep 8 (resp. 4)
// loads in flight while the previous group's WMMAs retire (double buffering).
#define LD8(buf, g)                                                     \
  { _Pragma("unroll") for (int u = 0; u < 8; ++u)                       \
      buf[u] = *(const v2f*)(brow + ((g) * 8 + u) * 4); }
#define MM8(buf, g)                                                     \
  { _Pragma("unroll") for (int u = 0; u < 8; ++u)                       \
      c = wmma_f32(af[(g) * 8 + u], buf[u], c); }

#define LD4(buf, g)                                                     \
  { _Pragma("unroll") for (int u = 0; u < 4; ++u)                       \
      buf[u] = *(const v2f*)(brow + ((g) * 4 + u) * 4); }
#define MM4D(buf, g)                                                    \
  { _Pragma("unroll") for (int u = 0; u < 4; ++u) {                     \
      c  = wmma_f32(afE[(g) * 4 + u], buf[u], c);                       \
      c2 = wmma_f32(afD[(g) * 4 + u], buf[u], c2); } }

// ---------------- kernel 0: sq[i] = ||e_i||^2 ------------------------------
__global__ void k_sq(const float* __restrict__ E, float* __restrict__ sq) {
  const int wave = threadIdx.x >> 5;
  const int lane = threadIdx.x & 31;
  const int row  = blockIdx.x * 4 + wave;
  float acc = 0.f;
#pragma unroll
  for (int q = 0; q < 4; ++q) {
    float v = E[row * D_N + lane + q * 32];
    acc += v * v;
  }
#pragma unroll
  for (int off = 16; off >= 1; off >>= 1)
    acc += __shfl_xor(acc, off, 32);
  if (lane == 0) sq[row] = acc;
}

// ---------------- kernel 1: per-row min_pos / max_neg ----------------------
__global__ void k_minmax(const float* __restrict__ E, const int* __restrict__ lab,
                         float* __restrict__ min_pos, float* __restrict__ max_neg) {
  const int i0   = blockIdx.x * 16;
  const int wave = threadIdx.x >> 5;
  const int lane = threadIdx.x & 31;
  const int half = lane >> 4;
  const int mr   = lane & 15;

  int labi[8];
#pragma unroll
  for (int r = 0; r < 8; ++r) labi[r] = lab[i0 + r + 8 * half];

  float minp[8], maxn[8];
#pragma unroll
  for (int r = 0; r < 8; ++r) { minp[r] = BIGF; maxn[r] = -BIGF; }

  // A fragments: constant over the whole column loop -> keep in registers
  const float* arow = E + (size_t)(i0 + mr) * D_N + 2 * half;
  v2f af[32];
#pragma unroll
  for (int kk = 0; kk < 32; ++kk) af[kk] = *(const v2f*)(arow + kk * 4);

  const int tend = wave * TPW + TPW;
  for (int t = wave * TPW; t < tend; ++t) {
    const int j0 = t * 16;
    const int labn = lab[j0 + mr];
    const float* brow = E + (size_t)(j0 + mr) * D_N + 2 * half;
    if (t + 1 < tend)
      __builtin_prefetch(E + (size_t)((t + 1) * 16 + mr) * D_N, 0, 0);
    v8f c = {0.f, 0.f, 0.f, 0.f, 0.f, 0.f, 0.f, 0.f};
    v2f bA[8], bB[8];
    LD8(bA, 0)
    LD8(bB, 1) MM8(bA, 0)
    LD8(bA, 2) MM8(bB, 1)
    LD8(bB, 3) MM8(bA, 2)
    MM8(bB, 3)
#pragma unroll
    for (int r = 0; r < 8; ++r) {
      float s = c[r];
      bool same = (labi[r] == labn);
      if (same && (s < ONE_M_EPS)) minp[r] = fminf(minp[r], s);
      if (!same)                   maxn[r] = fmaxf(maxn[r], s);
    }
  }
  // reduce across the 16 lanes of each half (rows live per half)
#pragma unroll
  for (int r = 0; r < 8; ++r) {
#pragma unroll
    for (int off = 8; off >= 1; off >>= 1) {
      minp[r] = fminf(minp[r], __shfl_xor(minp[r], off, 32));
      maxn[r] = fmaxf(maxn[r], __shfl_xor(maxn[r], off, 32));
    }
  }
  __shared__ float smin[WAVES][16];
  __shared__ float smax[WAVES][16];
  if (mr == 0) {
#pragma unroll
    for (int r = 0; r < 8; ++r) {
      smin[wave][r + 8 * half] = minp[r];
      smax[wave][r + 8 * half] = maxn[r];
    }
  }
  __syncthreads();
  if (threadIdx.x < 16) {
    float mn = BIGF, mx = -BIGF;
    for (int w = 0; w < WAVES; ++w) {
      mn = fminf(mn, smin[w][threadIdx.x]);
      mx = fmaxf(mx, smax[w][threadIdx.x]);
    }
    min_pos[i0 + threadIdx.x] = mn;
    max_neg[i0 + threadIdx.x] = mx;
  }
}

// ---------------- kernel 2: mining (argmin pos, pos_sum, valid) ------------
__global__ void k_mine(const float* __restrict__ E, const int* __restrict__ lab,
                       const float* __restrict__ min_pos, const float* __restrict__ max_neg,
                       int* __restrict__ jidx, float* __restrict__ pos_sum,
                       int* __restrict__ validf) {
  const int i0   = blockIdx.x * 16;
  const int wave = threadIdx.x >> 5;
  const int lane = threadIdx.x & 31;
  const int half = lane >> 4;
  const int mr   = lane & 15;

  int labi[8]; float mni[8], mxi[8];
#pragma unroll
  for (int r = 0; r < 8; ++r) {
    const int i = i0 + r + 8 * half;
    labi[r] = lab[i]; mni[r] = min_pos[i]; mxi[r] = max_neg[i];
  }

  float bv[8], ps[8]; int bi[8], fl[8];
#pragma unroll
  for (int r = 0; r < 8; ++r) { bv[r] = BIGF; bi[r] = 0x7fffffff; ps[r] = 0.f; fl[r] = 0; }

  const float* arow = E + (size_t)(i0 + mr) * D_N + 2 * half;
  v2f af[32];
#pragma unroll
  for (int kk = 0; kk < 32; ++kk) af[kk] = *(const v2f*)(arow + kk * 4);

  const int tend = wave * TPW + TPW;
  for (int t = wave * TPW; t < tend; ++t) {
    const int j0 = t * 16;
    const int labn = lab[j0 + mr];
    const float* brow = E + (size_t)(j0 + mr) * D_N + 2 * half;
    if (t + 1 < tend)
      __builtin_prefetch(E + (size_t)((t + 1) * 16 + mr) * D_N, 0, 0);
    v8f c = {0.f, 0.f, 0.f, 0.f, 0.f, 0.f, 0.f, 0.f};
    v2f bA[8], bB[8];
    LD8(bA, 0)
    LD8(bB, 1) MM8(bA, 0)
    LD8(bA, 2) MM8(bB, 1)
    LD8(bB, 3) MM8(bA, 2)
    MM8(bB, 3)
#pragma unroll
    for (int r = 0; r < 8; ++r) {
      float s = c[r];
      bool same = (labi[r] == labn);
      bool posk = same && (s < ONE_M_EPS) && ((s - MARGIN) < mxi[r]);
      bool negk = (!same) && ((s + MARGIN) > mni[r]);
      if (posk) {
        ps[r] += __expf(-SCALE_POS * (s - THRESH));
        fl[r] |= 1;
        int cand = j0 + mr;
        if (s < bv[r] || (s == bv[r] && cand < bi[r])) { bv[r] = s; bi[r] = cand; }
      }
      if (negk) fl[r] |= 2;
    }
  }
#pragma unroll
  for (int r = 0; r < 8; ++r) {
#pragma unroll
    for (int off = 8; off >= 1; off >>= 1) {
      float ov = __shfl_xor(bv[r], off, 32);
      int   oi = __shfl_xor(bi[r], off, 32);
      if (ov < bv[r] || (ov == bv[r] && oi < bi[r])) { bv[r] = ov; bi[r] = oi; }
      ps[r] += __shfl_xor(ps[r], off, 32);
      fl[r] |= __shfl_xor(fl[r], off, 32);
    }
  }
  __shared__ float s_bv[WAVES][16];
  __shared__ int   s_bi[WAVES][16];
  __shared__ float s_ps[WAVES][16];
  __shared__ int   s_fl[WAVES][16];
  if (mr == 0) {
#pragma unroll
    for (int r = 0; r < 8; ++r) {
      const int row16 = r + 8 * half;
      s_bv[wave][row16] = bv[r]; s_bi[wave][row16] = bi[r];
      s_ps[wave][row16] = ps[r]; s_fl[wave][row16] = fl[r];
    }
  }
  __syncthreads();
  if (threadIdx.x < 16) {
    float v = BIGF, p = 0.f; int ix = 0x7fffffff, f = 0;
    for (int w = 0; w < WAVES; ++w) {
      float ov = s_bv[w][threadIdx.x]; int oi = s_bi[w][threadIdx.x];
      if (ov < v || (ov == v && oi < ix)) { v = ov; ix = oi; }
      p += s_ps[w][threadIdx.x];
      f |= s_fl[w][threadIdx.x];
    }
    const int i = i0 + threadIdx.x;
    jidx[i]    = (v > 9.0e29f) ? 0 : ix;   // argmin over all-inf -> 0 (jnp semantics)
    pos_sum[i] = p;
    validf[i]  = (f == 3) ? 1 : 0;
  }
}

// ---------------- kernel 3: d_p, ||d_p|| (clamped), e_i . d_p --------------
__global__ void k_dprep(const float* __restrict__ E, const int* __restrict__ jidx,
                        float* __restrict__ dp, float* __restrict__ norm_p,
                        float* __restrict__ ed) {
  const int wave = threadIdx.x >> 5;
  const int lane = threadIdx.x & 31;
  const int row  = blockIdx.x * 4 + wave;
  const int jj   = jidx[row];
  float np2 = 0.f, e = 0.f;
#pragma unroll
  for (int q = 0; q < 4; ++q) {
    const int d = lane + q * 32;
    float ev = E[(size_t)row * D_N + d];
    float dv = E[(size_t)jj * D_N + d] - ev;
    dp[(size_t)row * D_N + d] = dv;
    np2 += dv * dv;
    e   += ev * dv;
  }
#pragma unroll
  for (int off = 16; off >= 1; off >>= 1) {
    np2 += __shfl_xor(np2, off, 32);
    e   += __shfl_xor(e,   off, 32);
  }
  if (lane == 0) {
    norm_p[row] = fmaxf(sqrtf(np2), COS_EPS);
    ed[row]     = e;
  }
}

// ---------------- kernel 4: mined-negative regularized sum -----------------
__global__ void k_neg(const float* __restrict__ E, const float* __restrict__ dp,
                      const int* __restrict__ lab, const float* __restrict__ min_pos,
                      const float* __restrict__ sq, const float* __restrict__ norm_p,
                      const float* __restrict__ ed, float* __restrict__ neg_sum) {
  const int i0   = blockIdx.x * 16;
  const int wave = threadIdx.x >> 5;
  const int lane = threadIdx.x & 31;
  const int half = lane >> 4;
  const int mr   = lane & 15;

  int labi[8]; float mni[8], sqi[8], edi[8], npi[8];
#pragma unroll
  for (int r = 0; r < 8; ++r) {
    const int i = i0 + r + 8 * half;
    labi[r] = lab[i]; mni[r] = min_pos[i]; sqi[r] = sq[i];
    edi[r] = ed[i];   npi[r] = norm_p[i];
  }

  float ns[8];
#pragma unroll
  for (int r = 0; r < 8; ++r) ns[r] = 0.f;

  // Two A-fragment sets (E rows and d_p rows), both loop-invariant.
  const float* arowE = E  + (size_t)(i0 + mr) * D_N + 2 * half;
  const float* arowD = dp + (size_t)(i0 + mr) * D_N + 2 * half;
  v2f afE[32], afD[32];
#pragma unroll
  for (int kk = 0; kk < 32; ++kk) afE[kk] = *(const v2f*)(arowE + kk * 4);
#pragma unroll
  for (int kk = 0; kk < 32; ++kk) afD[kk] = *(const v2f*)(arowD + kk * 4);

  const int tend = wave * TPW + TPW;
  for (int t = wave * TPW; t < tend; ++t) {
    const int j0 = t * 16;
    const int labn = lab[j0 + mr];
    const float sqn = sq[j0 + mr];
    const float* brow = E + (size_t)(j0 + mr) * D_N + 2 * half;
    if (t + 1 < tend)
      __builtin_prefetch(E + (size_t)((t + 1) * 16 + mr) * D_N, 0, 0);
    v8f c  = {0.f, 0.f, 0.f, 0.f, 0.f, 0.f, 0.f, 0.f};
    v8f c2 = {0.f, 0.f, 0.f, 0.f, 0.f, 0.f, 0.f, 0.f};
    v2f bA[4], bB[4];
    LD4(bA, 0)
    LD4(bB, 1) MM4D(bA, 0)
    LD4(bA, 2) MM4D(bB, 1)
    LD4(bB, 3) MM4D(bA, 2)
    LD4(bA, 4) MM4D(bB, 3)
    LD4(bB, 5) MM4D(bA, 4)
    LD4(bA, 6) MM4D(bB, 5)
    LD4(bB, 7) MM4D(bA, 6)
    MM4D(bB, 7)
#pragma unroll
    for (int r = 0; r < 8; ++r) {
      float s = c[r];
      if ((labi[r] != labn) && ((s + MARGIN) > mni[r])) {
        float dt    = c2[r] - edi[r];
        float nn    = sqrtf(fmaxf(sqn + sqi[r] - 2.f * s, 1e-12f));
        float denom = fmaxf(nn, COS_EPS) * npi[r];
        float reg   = GAMMA * dt / denom;
        ns[r] += __expf(SCALE_NEG * (s - THRESH - reg));
      }
    }
  }
#pragma unroll
  for (int r = 0; r < 8; ++r) {
#pragma unroll
    for (int off = 8; off >= 1; off >>= 1)
      ns[r] += __shfl_xor(ns[r], off, 32);
  }
  __shared__ float s_ns[WAVES][16];
  if (mr == 0) {
#pragma unroll
    for (int r = 0; r < 8; ++r) s_ns[wave][r + 8 * half] = ns[r];
  }
  __syncthreads();
  if (threadIdx.x < 16) {
    float acc = 0.f;
    for (int w = 0; w < WAVES; ++w) acc += s_ns[w][threadIdx.x];
    neg_sum[i0 + threadIdx.x] = acc;
  }
}

// ---------------- kernel 5: final scalar reduction -------------------------
__global__ void k_final(const float* __restrict__ pos_sum, const float* __restrict__ neg_sum,
                        const int* __restrict__ validf, float* __restrict__ out) {
  __shared__ float sacc[256];
  float acc = 0.f;
  for (int i = threadIdx.x; i < B_N; i += 256) {
    if (validf[i])
      acc += log1pf(pos_sum[i]) * (1.0f / SCALE_POS) +
             log1pf(neg_sum[i]) * (1.0f / SCALE_NEG);
  }
  sacc[threadIdx.x] = acc;
  __syncthreads();
  for (int s = 128; s >= 1; s >>= 1) {
    if (threadIdx.x < s) sacc[threadIdx.x] += sacc[threadIdx.x + s];
    __syncthreads();
  }
  if (threadIdx.x == 0) out[0] = sacc[0] / (float)B_N;
}

extern "C" void kernel_launch(void* const* d_in, const int* in_sizes, int n_in,
                              void* d_out, int out_size, void* d_ws, size_t ws_size,
                              hipStream_t stream) {
  (void)in_sizes; (void)n_in; (void)out_size; (void)ws_size;
  const float* E   = (const float*)d_in[0];
  const int*   lab = (const int*)d_in[1];
  float* out = (float*)d_out;

  float* ws      = (float*)d_ws;
  float* min_pos = ws + 0 * B_N;
  float* max_neg = ws + 1 * B_N;
  float* pos_sum = ws + 2 * B_N;
  float* neg_sum = ws + 3 * B_N;
  float* sq      = ws + 4 * B_N;
  float* norm_p  = ws + 5 * B_N;
  float* ed      = ws + 6 * B_N;
  int*   jidx    = (int*)(ws + 7 * B_N);
  int*   validf  = (int*)(ws + 8 * B_N);
  float* dp      = ws + 9 * B_N;            // B_N * D_N floats

  k_sq    <<<B_N / 4,  128, 0, stream>>>(E, sq);
  k_minmax<<<B_N / 16, 256, 0, stream>>>(E, lab, min_pos, max_neg);
  k_mine  <<<B_N / 16, 256, 0, stream>>>(E, lab, min_pos, max_neg, jidx, pos_sum, validf);
  k_dprep <<<B_N / 4,  128, 0, stream>>>(E, jidx, dp, norm_p, ed);
  k_neg   <<<B_N / 16, 256, 0, stream>>>(E, dp, lab, min_pos, sq, norm_p, ed, neg_sum);
  k_final <<<1,        256, 0, stream>>>(pos_sum, neg_sum, validf, out);
}